// TrainingLoss_46797963657468
// MI455X (gfx1250) — compile-verified
//
#include <hip/hip_runtime.h>
#include <hip/hip_bf16.h>

// TrainingLoss: mean_i [ logits[i,y_i] + log(exp(-J) + sum_k exp(-logits[i,k])) ]
// B = 262144 rows, C = 128 cols, f32. Bandwidth-bound: 128 MiB / 23.3 TB/s ~ 5.6us.
// Strategy:
//   Stage 1: one wave per row-group. 32 lanes x float4 = one full 512B row per
//            wave per iteration, perfectly coalesced global_load_b128. Default
//            (RT) cache hints so the 192MB L2 retains the 128MiB input across
//            graph replays. exp via fast v_exp_f32. Cross-lane butterfly
//            reduce (wave32!), per-block partial written to workspace.
//   Stage 2: single wave reduces the 1024 partials deterministically using
//            V_WMMA_F32_16X16X4_F32 with B = ones: D[m][n] = rowsum_m(A), so
//            sum(D) = 16 * sum(A). Accumulate C across iterations, fold 8
//            accumulator VGPRs + 32 lanes, divide by 16, scale by 1/B.

#define J_HYPER 0.01f
#define NBLK 1024
#define TPB 256
#define WAVES_PER_BLOCK (TPB / 32)

typedef __attribute__((ext_vector_type(2))) float v2f;
typedef __attribute__((ext_vector_type(8))) float v8f;

__global__ __launch_bounds__(TPB) void loss_stage1(
    const float* __restrict__ mo,   // [B,128] f32
    const int* __restrict__ tgt,    // [B] int32
    float* __restrict__ partial,    // [NBLK] f32
    int B)
{
    __shared__ float sred[WAVES_PER_BLOCK];

    const int lane   = threadIdx.x & 31;
    const int wid    = threadIdx.x >> 5;
    const int gwave  = blockIdx.x * WAVES_PER_BLOCK + wid;
    const int nwaves = gridDim.x * WAVES_PER_BLOCK;
    const float c0   = __expf(-J_HYPER);   // exp(-j), constant

    float acc = 0.0f;

    for (int row = gwave; row < B; row += nwaves) {
        // lane l owns columns [4l, 4l+4): wave covers the whole 512B row,
        // consecutive 16B per lane -> single coalesced b128 burst.
        const float4* rp = (const float4*)(mo + (size_t)row * 128);
        float4 v = rp[lane];

        float s = __expf(-v.x) + __expf(-v.y) + __expf(-v.z) + __expf(-v.w);

        int y = tgt[row];
        float g = 0.0f;
        if ((y >> 2) == lane) {
            g = (y & 2) ? ((y & 1) ? v.w : v.z)
                        : ((y & 1) ? v.y : v.x);
        }

        // wave32 butterfly: s -> full row exp-sum, g -> gathered logit
        #pragma unroll
        for (int off = 16; off >= 1; off >>= 1) {
            s += __shfl_xor(s, off, 32);
            g += __shfl_xor(g, off, 32);
        }

        acc += g + __logf(c0 + s);   // identical value in all 32 lanes
    }

    if (lane == 0) sred[wid] = acc;
    __syncthreads();

    if (threadIdx.x == 0) {
        float t = 0.0f;
        #pragma unroll
        for (int i = 0; i < WAVES_PER_BLOCK; ++i) t += sred[i];
        partial[blockIdx.x] = t;   // fixed slot -> deterministic
    }
}

__global__ __launch_bounds__(32) void loss_stage2(
    const float* __restrict__ partial, // [P], P % 64 == 0
    float* __restrict__ out,
    int P, float invB)
{
    const int lane = threadIdx.x;   // single wave, EXEC all-1s (WMMA requirement)

    v8f acc = {};                   // C/D accumulator, 16x16 f32
    v2f b;                          // B = ones(4x16): D[m][n] = sum_k A[m,k]
    b.x = 1.0f; b.y = 1.0f;

    for (int base = 0; base < P; base += 64) {
        // Pack 64 partials bijectively into the 16x4 f32 A layout (2 VGPRs x
        // 32 lanes). Exact (m,k) slot mapping is irrelevant for a global sum.
        v2f a;
        a.x = partial[base + 2 * lane];
        a.y = partial[base + 2 * lane + 1];
        acc = __builtin_amdgcn_wmma_f32_16x16x4_f32(
            /*neg_a=*/false, a, /*neg_b=*/false, b,
            /*c_mod=*/(short)0, acc, /*reuse_a=*/false, /*reuse_b=*/false);
    }

    // sum(D over all 256 elements) = 16 * sum(partials)
    float t = acc[0] + acc[1] + acc[2] + acc[3] +
              acc[4] + acc[5] + acc[6] + acc[7];
    #pragma unroll
    for (int off = 16; off >= 1; off >>= 1) t += __shfl_xor(t, off, 32);

    if (lane == 0) out[0] = t * (1.0f / 16.0f) * invB;
}

extern "C" void kernel_launch(void* const* d_in, const int* in_sizes, int n_in,
                              void* d_out, int out_size, void* d_ws, size_t ws_size,
                              hipStream_t stream) {
    const float* mo  = (const float*)d_in[0];
    const int*   tgt = (const int*)d_in[1];
    const int    B   = in_sizes[1];          // 262144

    float* partial = (float*)d_ws;           // NBLK floats = 4 KB scratch

    loss_stage1<<<NBLK, TPB, 0, stream>>>(mo, tgt, partial, B);
    loss_stage2<<<1, 32, 0, stream>>>(partial, (float*)d_out, NBLK,
                                      1.0f / (float)B);
}